// GINBlock_14860586844309
// MI455X (gfx1250) — compile-verified
//
#include <hip/hip_runtime.h>
#include <hip/hip_bf16.h>

typedef float v2f __attribute__((ext_vector_type(2)));
typedef float v8f __attribute__((ext_vector_type(8)));

#define D       128
#define D2      256
#define NGRAPHS 2000

// ---------------------------------------------------------------------------
// K1: agg = (1 + eps) * node_hidden      (float4 streaming)
// ---------------------------------------------------------------------------
__global__ void k_init_agg(const float* __restrict__ nh,
                           const float* __restrict__ eps,
                           float* __restrict__ agg, int total4) {
    int i = blockIdx.x * blockDim.x + threadIdx.x;
    if (i >= total4) return;
    float s = 1.0f + eps[0];
    float4 v = ((const float4*)nh)[i];
    float4 r;
    r.x = v.x * s; r.y = v.y * s; r.z = v.z * s; r.w = v.w * s;
    ((float4*)agg)[i] = r;
}

// ---------------------------------------------------------------------------
// K2: zero per-graph counts
// ---------------------------------------------------------------------------
__global__ void k_zero_counts(float* __restrict__ counts) {
    int i = blockIdx.x * blockDim.x + threadIdx.x;
    if (i < NGRAPHS) counts[i] = 0.0f;
}

// ---------------------------------------------------------------------------
// K3: counts[batch[i]] += 1
// ---------------------------------------------------------------------------
__global__ void k_counts(const int* __restrict__ batch,
                         float* __restrict__ counts, int N) {
    int i = blockIdx.x * blockDim.x + threadIdx.x;
    if (i >= N) return;
    atomicAdd(&counts[batch[i]], 1.0f);
}

// ---------------------------------------------------------------------------
// K4: edge scatter.  One wave32 per edge; lane handles 4 columns.
//     agg[dst] += edge_hidden[e] + node_hidden[src]   (global_atomic_add_f32)
// ---------------------------------------------------------------------------
__global__ __launch_bounds__(256)
void k_edges(const float* __restrict__ eh, const float* __restrict__ nh,
             const int* __restrict__ ei, float* __restrict__ agg, int E) {
    int e    = blockIdx.x * 8 + (threadIdx.x >> 5);
    int lane = threadIdx.x & 31;
    if (e >= E) return;
    int s = ei[e];        // row 0 of [2,E]
    int d = ei[E + e];    // row 1 of [2,E]
    float4 m = ((const float4*)eh)[e * (D / 4) + lane];
    float4 x = ((const float4*)nh)[s * (D / 4) + lane];
    float* p = agg + (size_t)d * D + lane * 4;
    atomicAdd(p + 0, m.x + x.x);
    atomicAdd(p + 1, m.y + x.y);
    atomicAdd(p + 2, m.z + x.z);
    atomicAdd(p + 3, m.w + x.w);
}

// ---------------------------------------------------------------------------
// K5: fused MLP (fp32 WMMA) + LayerNorm + GraphNorm + ReLU + residual.
// One wave32 per 16-row node tile.
//
// WMMA f32 16x16x4 operand layout (ISA 7.12.2):
//   A (16x4):  lane L -> m = L%16, k_base = 2*(L/16); vgpr j holds k_base+j
//   B (4x16):  lane L -> n = L%16, k_base = 2*(L/16); vgpr j holds B[k_base+j][n]
//   C/D:       vgpr r of lane L -> element [m = r + 8*(L/16)][n = L%16]
// ---------------------------------------------------------------------------
__global__ __launch_bounds__(32)
void k_mlp(const float* __restrict__ agg, const float* __restrict__ nh,
           const int* __restrict__ batch, const float* __restrict__ counts,
           const float* __restrict__ W1, const float* __restrict__ b1,
           const float* __restrict__ W2, const float* __restrict__ b2,
           const float* __restrict__ gamma, const float* __restrict__ beta,
           float* __restrict__ out, int N) {
    __shared__ __align__(16) float hA[16 * D];   // input tile, reused as output tile
    __shared__ __align__(16) float h2[16 * D2];  // post-ReLU intermediate

    const int lane = threadIdx.x;      // 0..31
    const int row0 = blockIdx.x * 16;
    const int half = lane >> 4;        // 0 or 1
    const int n    = lane & 15;        // column / A-row within tile
    const int m    = n;

    // ---- stage 16x128 agg tile into LDS (flat copy, zero-pad tail) ----
    {
        const float4* a4 = (const float4*)agg;
        float4* s4 = (float4*)hA;
        int base4  = row0 * (D / 4);
        int limit4 = N * (D / 4);
        for (int i = lane; i < 16 * D / 4; i += 32) {
            float4 v = (base4 + i < limit4) ? a4[base4 + i]
                                            : make_float4(0.f, 0.f, 0.f, 0.f);
            s4[i] = v;
        }
    }
    __syncthreads();

    // ---- GEMM1: [16,128] x [128,256] + b1, ReLU -> h2 ----
    for (int nt = 0; nt < 16; ++nt) {
        const float bv = b1[nt * 16 + n];
        v8f c;
#pragma unroll
        for (int r = 0; r < 8; ++r) c[r] = bv;     // bias depends on column only
        const float* hrow = &hA[m * D];
        const float* wcol = &W1[nt * 16 + n];
#pragma unroll 4
        for (int kk = 0; kk < 32; ++kk) {
            int k = kk * 4 + 2 * half;
            v2f a = *(const v2f*)(hrow + k);
            v2f b;
            b.x = wcol[(size_t)k * D2];
            b.y = wcol[(size_t)(k + 1) * D2];
            c = __builtin_amdgcn_wmma_f32_16x16x4_f32(false, a, false, b,
                                                      (short)0, c, false, false);
        }
#pragma unroll
        for (int r = 0; r < 8; ++r)
            h2[(r + 8 * half) * D2 + nt * 16 + n] = fmaxf(c[r], 0.0f);
    }
    __syncthreads();

    // ---- GEMM2: [16,256] x [256,128] + b2 -> hA (reused) ----
    for (int nt = 0; nt < 8; ++nt) {
        const float bv = b2[nt * 16 + n];
        v8f c;
#pragma unroll
        for (int r = 0; r < 8; ++r) c[r] = bv;
        const float* hrow = &h2[m * D2];
        const float* wcol = &W2[nt * 16 + n];
#pragma unroll 4
        for (int kk = 0; kk < 64; ++kk) {
            int k = kk * 4 + 2 * half;
            v2f a = *(const v2f*)(hrow + k);
            v2f b;
            b.x = wcol[(size_t)k * D];
            b.y = wcol[(size_t)(k + 1) * D];
            c = __builtin_amdgcn_wmma_f32_16x16x4_f32(false, a, false, b,
                                                      (short)0, c, false, false);
        }
#pragma unroll
        for (int r = 0; r < 8; ++r)
            hA[(r + 8 * half) * D + nt * 16 + n] = c[r];
    }
    __syncthreads();

    // ---- LayerNorm + GraphNorm + ReLU + residual (wave32 shuffle reductions) ----
    float4 gm = ((const float4*)gamma)[lane];
    float4 bt = ((const float4*)beta)[lane];
    for (int r = 0; r < 16; ++r) {
        int row = row0 + r;
        if (row >= N) break;                       // uniform across wave
        float4 x = ((const float4*)hA)[r * (D / 4) + lane];
        float s  = x.x + x.y + x.z + x.w;
        float s2 = x.x * x.x + x.y * x.y + x.z * x.z + x.w * x.w;
#pragma unroll
        for (int off = 16; off > 0; off >>= 1) {
            s  += __shfl_xor(s,  off, 32);
            s2 += __shfl_xor(s2, off, 32);
        }
        float mean = s * (1.0f / D);
        float var  = s2 * (1.0f / D) - mean * mean;
        float rsig = rsqrtf(var + 1e-5f);
        float gn   = rsqrtf(counts[batch[row]]);
        float4 nhv = ((const float4*)nh)[row * (D / 4) + lane];
        float4 o;
        o.x = fmaxf(((x.x - mean) * rsig * gm.x + bt.x) * gn, 0.0f) + nhv.x;
        o.y = fmaxf(((x.y - mean) * rsig * gm.y + bt.y) * gn, 0.0f) + nhv.y;
        o.z = fmaxf(((x.z - mean) * rsig * gm.z + bt.z) * gn, 0.0f) + nhv.z;
        o.w = fmaxf(((x.w - mean) * rsig * gm.w + bt.w) * gn, 0.0f) + nhv.w;
        ((float4*)out)[row * (D / 4) + lane] = o;
    }
}

// ---------------------------------------------------------------------------
extern "C" void kernel_launch(void* const* d_in, const int* in_sizes, int n_in,
                              void* d_out, int out_size, void* d_ws, size_t ws_size,
                              hipStream_t stream) {
    const float* node_hidden = (const float*)d_in[0];
    const int*   edge_index  = (const int*)d_in[1];
    const float* edge_hidden = (const float*)d_in[2];
    const int*   batch       = (const int*)d_in[3];
    const float* W1    = (const float*)d_in[4];
    const float* b1    = (const float*)d_in[5];
    const float* W2    = (const float*)d_in[6];
    const float* b2    = (const float*)d_in[7];
    const float* eps   = (const float*)d_in[8];
    const float* gamma = (const float*)d_in[9];
    const float* beta  = (const float*)d_in[10];

    const int N = in_sizes[0] / D;     // 100000
    const int E = in_sizes[1] / 2;     // 600000

    float* agg    = (float*)d_ws;                  // N*D floats
    float* counts = agg + (size_t)N * D;           // NGRAPHS floats
    float* out    = (float*)d_out;

    int total4 = N * (D / 4);
    k_init_agg<<<(total4 + 255) / 256, 256, 0, stream>>>(node_hidden, eps, agg, total4);
    k_zero_counts<<<(NGRAPHS + 255) / 256, 256, 0, stream>>>(counts);
    k_counts<<<(N + 255) / 256, 256, 0, stream>>>(batch, counts, N);
    k_edges<<<(E + 7) / 8, 256, 0, stream>>>(edge_hidden, node_hidden, edge_index, agg, E);
    k_mlp<<<(N + 15) / 16, 32, 0, stream>>>(agg, node_hidden, batch, counts,
                                            W1, b1, W2, b2, gamma, beta, out, N);
}